// GCN_25795573579864
// MI455X (gfx1250) — compile-verified
//
#include <hip/hip_runtime.h>
#include <hip/hip_bf16.h>

// GCN layer: out = PReLU( adj @ (seq @ W^T) + b )
// B=1, N=16384, IN_FT=128, OUT_FT=64, all fp32.
// Exact fp32 math via V_WMMA_F32_16X16X4_F32 (wave32 CDNA5 WMMA).
//
// adj (1 GiB) is streamed once from HBM (bandwidth roof ~46 us @ 23.3 TB/s,
// AI = 32 FLOP/B -> near the f32-WMMA compute roof). seq_fts (4 MB) lives in
// L2 and is double-buffer staged into LDS per block.

typedef __attribute__((ext_vector_type(2))) float v2f;
typedef __attribute__((ext_vector_type(8))) float v8f;

static constexpr int N_NODES = 16384;
static constexpr int IN_FT   = 128;
static constexpr int OUT_FT  = 64;

static constexpr int KC      = 32;          // K-chunk staged per LDS buffer
static constexpr int BPAD    = KC + 4;      // 36: conflict-free ds_load_b64 reads
static constexpr int KHALF   = N_NODES / 2; // per-team K range (8192)
static constexpr int NCHUNK  = KHALF / KC;  // 256 chunks per team
static constexpr int BUFSZ   = OUT_FT * BPAD;

static __device__ __forceinline__ v8f wmma_f32(v2f a, v2f b, v8f c) {
    // 8 args: (neg_a, A, neg_b, B, c_mod, C, reuse_a, reuse_b)
    return __builtin_amdgcn_wmma_f32_16x16x4_f32(
        false, a, false, b, (short)0, c, false, false);
}

// ---------------------------------------------------------------------------
// Kernel 1: seq_fts[n, o] = sum_i seq[n, i] * W[o, i]
// One wave per 16-row strip; 4 N-tiles, K=128. Tiny (0.27 GFLOP).
// ---------------------------------------------------------------------------
__global__ __launch_bounds__(32) void gcn_fc_kernel(
    const float* __restrict__ seq, const float* __restrict__ W,
    float* __restrict__ sf)
{
    const int lane = threadIdx.x;
    const int m    = lane & 15;
    const int khi  = lane >> 4;

    const int r0 = blockIdx.x * 16;

    v8f c0 = {}, c1 = {}, c2 = {}, c3 = {};

    const float* arow = seq + (size_t)(r0 + m) * IN_FT + 2 * khi;
    const float* wrow = W + (size_t)m * IN_FT + 2 * khi;   // B[k,n] = W[n,k]

    #pragma unroll 4
    for (int kb = 0; kb < IN_FT; kb += 4) {
        v2f a  = *(const v2f*)(arow + kb);
        v2f b0 = *(const v2f*)(wrow + kb);
        v2f b1 = *(const v2f*)(wrow + kb + 16 * IN_FT);
        v2f b2 = *(const v2f*)(wrow + kb + 32 * IN_FT);
        v2f b3 = *(const v2f*)(wrow + kb + 48 * IN_FT);
        c0 = wmma_f32(a, b0, c0);
        c1 = wmma_f32(a, b1, c1);
        c2 = wmma_f32(a, b2, c2);
        c3 = wmma_f32(a, b3, c3);
    }

    float* orow = sf + (size_t)(r0 + 8 * khi) * OUT_FT + m;
    #pragma unroll
    for (int r = 0; r < 8; ++r) {
        orow[(size_t)r * OUT_FT +  0] = c0[r];
        orow[(size_t)r * OUT_FT + 16] = c1[r];
        orow[(size_t)r * OUT_FT + 32] = c2[r];
        orow[(size_t)r * OUT_FT + 48] = c3[r];
    }
}

// ---------------------------------------------------------------------------
// Kernel 2: out = PReLU( adj @ seq_fts + b )
// Block = 256 threads = 8 waves, 64 rows. Waves 0-3 (team 0) take K in
// [0,8192), waves 4-7 (team 1) take [8192,16384): 2048 waves machine-wide,
// no duplicated HBM traffic. Each team double-buffers its seq_fts chunk in
// LDS (prefetch next chunk during WMMA on current). Final cross-team add via
// LDS, fused bias + PReLU epilogue by team 0.
// ---------------------------------------------------------------------------
__global__ __launch_bounds__(256) void gcn_agg_kernel(
    const float* __restrict__ adj, const float* __restrict__ sf,
    const float* __restrict__ bias, const float* __restrict__ prelu_a,
    float* __restrict__ out)
{
    __shared__ __align__(16) float Bs[4 * BUFSZ];   // [team][buf] = 36.9 KB

    const int tid  = threadIdx.x;        // 0..255
    const int lane = tid & 31;
    const int wave = tid >> 5;           // 0..7
    const int team = wave >> 2;          // 0..1 -> K half
    const int w4   = wave & 3;           // row strip within block
    const int m    = lane & 15;
    const int khi  = lane >> 4;          // WMMA K-pair half

    const int r0    = blockIdx.x * 64 + w4 * 16;
    const int kbase = team * KHALF;

    v8f c0 = {}, c1 = {}, c2 = {}, c3 = {};

    const float* arow = adj + (size_t)(r0 + m) * N_NODES + kbase + 2 * khi;

    // Staging role (per 128-thread team): 4 threads per k-row, 16 floats each.
    const int ltid = tid & 127;
    const int krow = ltid >> 2;          // 0..31
    const int n0   = (ltid & 3) * 16;    // 0,16,32,48
    const float* ssrc = sf + (size_t)(kbase + krow) * OUT_FT + n0;
    float* buf0 = &Bs[(team * 2 + 0) * BUFSZ];
    float* buf1 = &Bs[(team * 2 + 1) * BUFSZ];

    // --- prologue: stage chunk 0 into buf0 (transposed: Bs[n][k]) ---
    {
        #pragma unroll
        for (int j = 0; j < 4; ++j) {
            float4 v = *(const float4*)(ssrc + 4 * j);
            const int n = n0 + 4 * j;
            buf0[(n + 0) * BPAD + krow] = v.x;
            buf0[(n + 1) * BPAD + krow] = v.y;
            buf0[(n + 2) * BPAD + krow] = v.z;
            buf0[(n + 3) * BPAD + krow] = v.w;
        }
    }
    __syncthreads();

    // --- main loop: compute chunk i from buf[i&1], prefetch i+1 into regs,
    //     drain to the other buffer; one barrier per chunk. ---
    for (int i = 0; i < NCHUNK; ++i) {
        float4 pf0, pf1, pf2, pf3;
        const bool have_next = (i + 1 < NCHUNK);
        if (have_next) {
            const float* s2 = ssrc + (size_t)(i + 1) * KC * OUT_FT;
            pf0 = *(const float4*)(s2 +  0);
            pf1 = *(const float4*)(s2 +  4);
            pf2 = *(const float4*)(s2 +  8);
            pf3 = *(const float4*)(s2 + 12);
        }

        const float* bufc = (i & 1) ? buf1 : buf0;
        const float* ap   = arow + (size_t)i * KC;
        #pragma unroll
        for (int ks = 0; ks < KC; ks += 4) {
            v2f a = *(const v2f*)(ap + ks);
            const int kk = ks + 2 * khi;
            const float* bp = bufc + m * BPAD + kk;
            v2f b0 = *(const v2f*)(bp);
            v2f b1 = *(const v2f*)(bp + 16 * BPAD);
            v2f b2 = *(const v2f*)(bp + 32 * BPAD);
            v2f b3 = *(const v2f*)(bp + 48 * BPAD);
            c0 = wmma_f32(a, b0, c0);
            c1 = wmma_f32(a, b1, c1);
            c2 = wmma_f32(a, b2, c2);
            c3 = wmma_f32(a, b3, c3);
        }

        if (have_next) {
            float* dst = (i & 1) ? buf0 : buf1;
            const float4 v[4] = {pf0, pf1, pf2, pf3};
            #pragma unroll
            for (int j = 0; j < 4; ++j) {
                const int n = n0 + 4 * j;
                dst[(n + 0) * BPAD + krow] = v[j].x;
                dst[(n + 1) * BPAD + krow] = v[j].y;
                dst[(n + 2) * BPAD + krow] = v[j].z;
                dst[(n + 3) * BPAD + krow] = v[j].w;
            }
        }
        __syncthreads();
    }

    // --- cross-team reduction (reuse Bs as 16 KB scratch) ---
    if (team == 1) {
        #pragma unroll
        for (int r = 0; r < 8; ++r) {
            Bs[(((w4 * 4 + 0) * 8 + r) << 5) + lane] = c0[r];
            Bs[(((w4 * 4 + 1) * 8 + r) << 5) + lane] = c1[r];
            Bs[(((w4 * 4 + 2) * 8 + r) << 5) + lane] = c2[r];
            Bs[(((w4 * 4 + 3) * 8 + r) << 5) + lane] = c3[r];
        }
    }
    __syncthreads();
    if (team == 0) {
        #pragma unroll
        for (int r = 0; r < 8; ++r) {
            c0[r] += Bs[(((w4 * 4 + 0) * 8 + r) << 5) + lane];
            c1[r] += Bs[(((w4 * 4 + 1) * 8 + r) << 5) + lane];
            c2[r] += Bs[(((w4 * 4 + 2) * 8 + r) << 5) + lane];
            c3[r] += Bs[(((w4 * 4 + 3) * 8 + r) << 5) + lane];
        }

        const float a_slope = prelu_a[0];
        const float bv0 = bias[m +  0];
        const float bv1 = bias[m + 16];
        const float bv2 = bias[m + 32];
        const float bv3 = bias[m + 48];

        float* orow = out + (size_t)(r0 + 8 * khi) * OUT_FT + m;
        #pragma unroll
        for (int r = 0; r < 8; ++r) {
            float v0 = c0[r] + bv0;
            float v1 = c1[r] + bv1;
            float v2 = c2[r] + bv2;
            float v3 = c3[r] + bv3;
            v0 = v0 > 0.0f ? v0 : a_slope * v0;
            v1 = v1 > 0.0f ? v1 : a_slope * v1;
            v2 = v2 > 0.0f ? v2 : a_slope * v2;
            v3 = v3 > 0.0f ? v3 : a_slope * v3;
            orow[(size_t)r * OUT_FT +  0] = v0;
            orow[(size_t)r * OUT_FT + 16] = v1;
            orow[(size_t)r * OUT_FT + 32] = v2;
            orow[(size_t)r * OUT_FT + 48] = v3;
        }
    }
}

extern "C" void kernel_launch(void* const* d_in, const int* in_sizes, int n_in,
                              void* d_out, int out_size, void* d_ws, size_t ws_size,
                              hipStream_t stream) {
    (void)in_sizes; (void)n_in; (void)out_size; (void)ws_size;
    const float* seq = (const float*)d_in[0];   // [1, 16384, 128]
    const float* adj = (const float*)d_in[1];   // [1, 16384, 16384]
    const float* W   = (const float*)d_in[2];   // [64, 128]
    const float* b   = (const float*)d_in[3];   // [64]
    const float* pa  = (const float*)d_in[4];   // [1]
    float* out = (float*)d_out;                 // [1, 16384, 64]
    float* sf  = (float*)d_ws;                  // seq_fts scratch: 4 MB

    gcn_fc_kernel<<<N_NODES / 16, 32, 0, stream>>>(seq, W, sf);
    gcn_agg_kernel<<<N_NODES / 64, 256, 0, stream>>>(adj, sf, b, pa, out);
}